// VoronoiTransform_63548336111964
// MI455X (gfx1250) — compile-verified
//
#include <hip/hip_runtime.h>
#include <math.h>

// Shapes from the reference: B=128, N=64, K=256, D=64 (all fp32)
#define BB 128
#define NN 64
#define KK 256
#define DD 64

typedef __attribute__((ext_vector_type(2))) float v2f;
typedef __attribute__((ext_vector_type(8))) float v8f;

__device__ __forceinline__ float softplus_f(float v) {
    // jax.nn.softplus: log1p(exp(-|x|)) + max(x, 0)
    return log1pf(expf(-fabsf(v))) + fmaxf(v, 0.0f);
}

__global__ __launch_bounds__(256)
void voronoi_fused_kernel(const float* __restrict__ x,
                          const float* __restrict__ anchor_raw,
                          const float* __restrict__ box_raw,
                          float* __restrict__ out)
{
    __shared__ float Ps[KK * DD];      // pts (64 KB)
    __shared__ float Xs[BB * DD];      // x rows for this n (32 KB)
    __shared__ float Dxs[BB * DD];     // del_x (32 KB)
    __shared__ float pqS[KK];          // |p_k|^2
    __shared__ float bmaxS[DD];
    __shared__ float bminS[DD];
    __shared__ int   nearS[BB];
    __shared__ float qBs[BB];          // |x_k|^2 per b
    __shared__ float cBs[BB];          // x_k . del_x per b
    __shared__ int   actS[BB];
    __shared__ float vminS[BB];
    __shared__ int   bndS[BB];

    const int n    = blockIdx.x;
    const int tid  = threadIdx.x;
    const int lane = tid & 31;
    const int wave = tid >> 5;
    const int half = lane >> 4;    // 0: lanes 0-15, 1: lanes 16-31
    const int l15  = lane & 15;
    const int jb   = half * 2;     // K-sub-offset inside a 4-wide f32 WMMA block
    const float INF_F = __builtin_inff();

    // ---- box constraints ----
    if (tid < DD) {
        float r0 = box_raw[(n * DD + tid) * 2 + 0];
        float r1 = box_raw[(n * DD + tid) * 2 + 1];
        bmaxS[tid] =  softplus_f(r0) + 1.0f;
        bminS[tid] = -(softplus_f(r1) + 1.0f);
    }
    __syncthreads();

    // ---- pts = (softsign(anchor)+1)/2 * (bmax-bmin) + bmin ----
    for (int idx = tid; idx < KK * DD; idx += 256) {
        int k = idx >> 6, d = idx & 63;
        float a = anchor_raw[(n * KK + k) * DD + d];
        float t = (a / (1.0f + fabsf(a)) + 1.0f) * 0.5f;
        Ps[idx] = t * (bmaxS[d] - bminS[d]) + bminS[d];
    }
    // ---- stage x rows for this n ----
    for (int idx = tid; idx < BB * DD; idx += 256) {
        int b = idx >> 6, d = idx & 63;
        Xs[idx] = x[(b * NN + n) * DD + d];
    }
    __syncthreads();

    // ---- |p_k|^2 ----
    if (tid < KK) {
        float s = 0.0f;
        for (int d = 0; d < DD; ++d) { float p = Ps[tid * DD + d]; s += p * p; }
        pqS[tid] = s;
    }
    __syncthreads();

    // ===== Phase 1: nearest anchor via WMMA  G = X . P^T  (argmin of pq - 2G) =====
    {
        const int b0 = wave * 16;                  // 8 waves x 16 rows = 128 b's
        float minv[8]; int mini[8];
        #pragma unroll
        for (int r = 0; r < 8; ++r) { minv[r] = INF_F; mini[r] = 0; }

        for (int kt = 0; kt < KK / 16; ++kt) {
            const int k0 = kt * 16;
            v8f acc = {};
            #pragma unroll
            for (int d0 = 0; d0 < DD; d0 += 4) {
                v2f a, bf;
                // A 16x4: lane holds row M=l15, K-cols jb, jb+1
                a.x  = Xs[(b0 + l15) * DD + d0 + jb];
                a.y  = Xs[(b0 + l15) * DD + d0 + jb + 1];
                // B 4x16 = (P^T) slice: B[j][c] = P[k0+c][d0+j]
                bf.x = Ps[(k0 + l15) * DD + d0 + jb];
                bf.y = Ps[(k0 + l15) * DD + d0 + jb + 1];
                acc = __builtin_amdgcn_wmma_f32_16x16x4_f32(
                        false, a, false, bf, (short)0, acc, false, false);
            }
            const int   kcol = k0 + l15;
            const float pq2  = pqS[kcol];
            #pragma unroll
            for (int r = 0; r < 8; ++r) {          // C VGPR r = row (r + 8*half)
                float s = pq2 - 2.0f * acc[r];
                if (s < minv[r]) { minv[r] = s; mini[r] = kcol; }
            }
        }
        // reduce across the 16 lanes of each half (b fixed, k varies across lanes)
        #pragma unroll
        for (int m = 1; m < 16; m <<= 1) {
            #pragma unroll
            for (int r = 0; r < 8; ++r) {
                float ov = __shfl_xor(minv[r], m, 32);
                int   oi = __shfl_xor(mini[r], m, 32);
                if (ov < minv[r] || (ov == minv[r] && oi < mini[r])) {
                    minv[r] = ov; mini[r] = oi;
                }
            }
        }
        if (l15 == 0) {
            #pragma unroll
            for (int r = 0; r < 8; ++r) nearS[b0 + r + half * 8] = mini[r];
        }
    }
    __syncthreads();

    // ---- per-b: del_x (unit), qB = |x_k|^2, cB = x_k . del_x ----
    if (tid < BB) {
        int b = tid, nb = nearS[b];
        float s = 0.0f;
        for (int d = 0; d < DD; ++d) {
            float dx = Xs[b * DD + d] - Ps[nb * DD + d];
            s += dx * dx;
        }
        float inv = 1.0f / sqrtf(s);
        float c = 0.0f;
        for (int d = 0; d < DD; ++d) {
            float del = (Xs[b * DD + d] - Ps[nb * DD + d]) * inv;
            Dxs[b * DD + d] = del;
            c += Ps[nb * DD + d] * del;
        }
        cBs[b] = c;
        qBs[b] = pqS[nb];
    }
    __syncthreads();

    // ===== Phase 2: Voronoi lambdas via two fused WMMA GEMMs (A=pts reused) =====
    // M1[k][b] = p_k . x_k(b),  M2[k][b] = p_k . del_x(b)
    {
        const int   b0    = wave * 16;             // 8 waves x 16 cols = 128 b's
        const int   bcol  = b0 + l15;
        const int   nbcol = nearS[bcol];
        const float qb    = qBs[bcol];
        const float cb    = cBs[bcol];
        float bestv = INF_F; int besti = 0; int bnd = 0;

        for (int kt = 0; kt < KK / 16; ++kt) {
            const int k0 = kt * 16;
            v8f acc1 = {}, acc2 = {};
            #pragma unroll
            for (int d0 = 0; d0 < DD; d0 += 4) {
                v2f a, b1, b2;
                a.x  = Ps[(k0 + l15) * DD + d0 + jb];       // A row = k0+l15
                a.y  = Ps[(k0 + l15) * DD + d0 + jb + 1];
                b1.x = Ps[nbcol * DD + d0 + jb];            // B1 col b -> x_k row
                b1.y = Ps[nbcol * DD + d0 + jb + 1];
                b2.x = Dxs[bcol * DD + d0 + jb];            // B2 col b -> del_x row
                b2.y = Dxs[bcol * DD + d0 + jb + 1];
                acc1 = __builtin_amdgcn_wmma_f32_16x16x4_f32(
                         false, a, false, b1, (short)0, acc1, false, false);
                acc2 = __builtin_amdgcn_wmma_f32_16x16x4_f32(
                         false, a, false, b2, (short)0, acc2, false, false);
            }
            #pragma unroll
            for (int r = 0; r < 8; ++r) {
                int   k    = k0 + r + half * 8;             // C row = k index
                float num  = pqS[k] + qb - 2.0f * acc1[r];  // |p_k - x_k|^2
                float den  = 2.0f * (acc2[r] - cb);         // 2 (p_k - x_k).del_x
                float lamb = (den == 0.0f) ? 0.0f : num / den;
                if (lamb > 0.0f && k != nbcol) bnd = 1;
                float cand = (lamb > 0.0f && k != nbcol) ? lamb : INF_F;
                if (cand < bestv || (cand == bestv && k < besti)) {
                    bestv = cand; besti = k;
                }
            }
        }
        // partner lane (other half) holds the other 8 rows of each k-tile
        {
            float ov = __shfl_xor(bestv, 16, 32);
            int   oi = __shfl_xor(besti, 16, 32);
            int   ob = __shfl_xor(bnd,   16, 32);
            if (ov < bestv || (ov == bestv && oi < besti)) { bestv = ov; besti = oi; }
            bnd |= ob;
        }
        if (half == 0) {
            actS[bcol]  = besti;
            vminS[bcol] = bestv;
            bndS[bcol]  = bnd;
        }
    }
    __syncthreads();

    // ===== Phase 3: box constraints, constraint selection, alpha, output =====
    if (tid < BB) {
        const int   b  = tid, nb = nearS[b];
        const float qb = qBs[b], cb = cBs[b];

        // box lambdas: entries [0,D) = (bmax - x_k)/del, [D,2D) = (x_k - bmin)/(-del)
        float bmv = INF_F; int bmi = 0;
        for (int d = 0; d < DD; ++d) {
            float xk  = Ps[nb * DD + d], del = Dxs[b * DD + d];
            float l1  = (bmaxS[d] - xk) / del;
            if (!(l1 >= 0.0f)) l1 = INF_F;              // NaN or negative -> INF
            if (l1 < bmv) { bmv = l1; bmi = d; }
        }
        for (int d = 0; d < DD; ++d) {
            float xk  = Ps[nb * DD + d], del = Dxs[b * DD + d];
            float l2  = (xk - bminS[d]) / (-del);
            if (!(l2 >= 0.0f)) l2 = INF_F;
            if (l2 < bmv) { bmv = l2; bmi = DD + d; }
        }

        const int  is_b    = bndS[b];
        const float vmin   = vminS[b];
        const bool use_box = (!is_b) || (bmv < vmin);

        float bf, a_xk, a_dx;
        if (use_box) {
            int   dim = (bmi < DD) ? bmi : (bmi - DD);
            float sgn = (bmi < DD) ? 1.0f : -1.0f;
            bf   = (bmi < DD) ? bmaxS[dim] : -bminS[dim];
            a_xk = sgn * Ps[nb * DD + dim];
            a_dx = sgn * Dxs[b * DD + dim];
        } else {
            int   kv = actS[b];
            float d1 = 0.0f, d2 = 0.0f;
            for (int d = 0; d < DD; ++d) {
                float pv = Ps[kv * DD + d];
                d1 += pv * Ps[nb * DD + d];
                d2 += pv * Dxs[b * DD + d];
            }
            bf   = pqS[kv] - qb;                        // |p|^2 - |x_k|^2
            a_xk = 2.0f * (d1 - qb);                    // a_v . x_k
            a_dx = 2.0f * (d2 - cb);                    // a_v . del_x
        }

        float lamb = (bf - a_xk) / a_dx;

        // alpha = nanmean over valid dims of (x - x_k) / (lamb * del)
        float ssum = 0.0f; int cnt = 0;
        for (int d = 0; d < DD; ++d) {
            float denom = lamb * Dxs[b * DD + d];
            if (denom != 0.0f) {                        // NaN != 0 is true (matches jnp)
                ssum += (Xs[b * DD + d] - Ps[nb * DD + d]) / denom;
                ++cnt;
            }
        }
        float alpha = ssum / fmaxf((float)cnt, 1.0f);

        for (int d = 0; d < DD; ++d) {
            float xk = Ps[nb * DD + d];
            out[(b * NN + n) * DD + d] = xk + alpha * (lamb * Dxs[b * DD + d]);
        }
    }
}

extern "C" void kernel_launch(void* const* d_in, const int* in_sizes, int n_in,
                              void* d_out, int out_size, void* d_ws, size_t ws_size,
                              hipStream_t stream) {
    (void)in_sizes; (void)n_in; (void)d_ws; (void)ws_size; (void)out_size;
    const float* x      = (const float*)d_in[0];
    const float* anchor = (const float*)d_in[1];
    const float* box    = (const float*)d_in[2];
    float*       out    = (float*)d_out;
    hipLaunchKernelGGL(voronoi_fused_kernel, dim3(NN), dim3(256), 0, stream,
                       x, anchor, box, out);
}